// DTMWeightLayer_63531156242954
// MI455X (gfx1250) — compile-verified
//
#include <hip/hip_runtime.h>

// DTM (distance-to-measure) layer for MI455X / gfx1250.
//  B=2, M=2048 points (2-D), N=6561 grid points, wB = 0.3 * sum(weight).
//
// Phase 1: d^2 matrix tiles via V_WMMA_F32_16X16X4_F32 using homogeneous
//          coordinates: A[g]=(g0,g1,|g|^2,1), B[:,m]=(-2x0,-2x1,1,|x|^2),
//          A*B = |g-x|^2 exactly (K=4 == the f32 WMMA K). Clamp-to-0 done
//          with a single inline-asm v_max_num_f32 (no canonicalize dup).
// Phase 2: exact weighted quantile by 31-step bitwise binary search over the
//          float bits of d^2 (monotone for d^2>=0). The 16x2048 d^2 block
//          lives in CDNA5's 320KB LDS (read as ds_load_b128); the per-lane
//          weight slice (64 floats) is cached in VGPRs so the hot loop only
//          streams d^2 from LDS.

typedef float v2f __attribute__((ext_vector_type(2)));
typedef float v8f __attribute__((ext_vector_type(8)));

#define M_PTS     2048
#define N_GRID    6561
#define ROWSTRIDE 2064              // 2048 + 16 pad: kills LDS bank conflicts, 16B aligned
#define NTILES_M  (M_PTS / 16)      // 128 point tiles
#define WAVES     16
#define BLOCK     (WAVES * 32)      // 512 threads, wave32
#define CHUNKS    (M_PTS / 4 / 32)  // 16 float4 per lane per row

__device__ __forceinline__ float wave_sum(float x) {
#pragma unroll
  for (int m = 16; m >= 1; m >>= 1) x += __shfl_xor(x, m, 32);
  return x;                         // full butterfly: all lanes hold the sum
}

__device__ __forceinline__ float clamp0(float x) {
  float r;
  asm("v_max_num_f32 %0, 0, %1" : "=v"(r) : "v"(x));  // single VALU op
  return r;
}

__global__ __launch_bounds__(BLOCK, 1)
void dtm_kernel(const float* __restrict__ pts,   // [B, M, 2]
                const float* __restrict__ wgt,   // [B, M]
                const float* __restrict__ grid,  // [N, 2]
                float* __restrict__ out)         // [B, N]
{
  extern __shared__ float smem[];
  float* ldsD = smem;                   // [16][ROWSTRIDE] d^2 block
  float* ldsW = smem + 16 * ROWSTRIDE;  // [M] weights

  const int b     = blockIdx.y;
  const int gbase = blockIdx.x * 16;
  const int tid   = (int)threadIdx.x;
  const int wave  = tid >> 5;
  const int lane  = tid & 31;
  const int half  = lane >> 4;          // 0: lanes 0-15, 1: lanes 16-31
  const int l16   = lane & 15;

  // ---- stage weights into LDS ----
  for (int i = tid; i < M_PTS; i += BLOCK) ldsW[i] = wgt[b * M_PTS + i];

  // ---- A fragment (grid tile), 16x4 f32: lanes 0-15 hold K=0,1; 16-31 K=2,3
  int grow = gbase + l16; if (grow > N_GRID - 1) grow = N_GRID - 1;
  const float gx = grid[2 * grow + 0];
  const float gy = grid[2 * grow + 1];
  v2f a;
  if (half == 0) { a.x = gx;             a.y = gy;   }   // K=0 -> g0, K=1 -> g1
  else           { a.x = gx*gx + gy*gy;  a.y = 1.0f; }   // K=2 -> |g|^2, K=3 -> 1

  // ---- phase 1: each wave produces 8 of the 128 16x16 d^2 tiles ----
#pragma unroll
  for (int i = 0; i < NTILES_M / WAVES; ++i) {
    const int tbase = (wave * (NTILES_M / WAVES) + i) * 16;
    const int m     = tbase + l16;
    const float px  = pts[(b * M_PTS + m) * 2 + 0];
    const float py  = pts[(b * M_PTS + m) * 2 + 1];
    v2f bb;                                              // B 4x16 f32 fragment
    if (half == 0) { bb.x = -2.0f * px;  bb.y = -2.0f * py;     }
    else           { bb.x = 1.0f;        bb.y = px*px + py*py;  }
    v8f c = {0.f,0.f,0.f,0.f,0.f,0.f,0.f,0.f};
    c = __builtin_amdgcn_wmma_f32_16x16x4_f32(false, a, false, bb,
                                              (short)0, c, false, false);
    // C/D layout: VGPR v -> row v (lanes 0-15) / row v+8 (lanes 16-31), col=l16
#pragma unroll
    for (int v = 0; v < 8; ++v)
      ldsD[(v + 8 * half) * ROWSTRIDE + tbase + l16] = clamp0(c[v]);
  }
  __syncthreads();

  // ---- phase 2: one wave per grid row, weighted quantile from LDS ----
  const int grow2 = gbase + wave;
  if (grow2 < N_GRID) {
    const float4* w4    = (const float4*)ldsW;                       // [512]
    const float4* rowD4 = (const float4*)(ldsD + wave * ROWSTRIDE);  // [512]

    // cache this lane's weight slice in VGPRs (loop-invariant over the search)
    float4 wr[CHUNKS];
#pragma unroll
    for (int i = 0; i < CHUNKS; ++i) wr[i] = w4[lane + 32 * i];

    // total weight -> mass bound wB (pure VALU from the register cache)
    float tw = 0.f;
#pragma unroll
    for (int i = 0; i < CHUNKS; ++i)
      tw += (wr[i].x + wr[i].y) + (wr[i].z + wr[i].w);
    tw = wave_sum(tw);
    const float wB = 0.3f * tw;

    // smallest float v (by bits) with  sum{w : d2 <= v} >= wB
    unsigned lo = 0u, hi = 0x7F800000u;                  // [0, +inf]
#pragma unroll 1
    for (int it = 0; it < 31; ++it) {
      const unsigned mid = (lo + hi) >> 1;
      const float t = __uint_as_float(mid);
      float s = 0.f;
#pragma unroll
      for (int i = 0; i < CHUNKS; ++i) {
        const float4 d = rowD4[lane + 32 * i];           // ds_load_b128
        s += ((d.x <= t) ? wr[i].x : 0.f) + ((d.y <= t) ? wr[i].y : 0.f)
           + ((d.z <= t) ? wr[i].z : 0.f) + ((d.w <= t) ? wr[i].w : 0.f);
      }
      s = wave_sum(s);                                   // uniform across wave
      if (s >= wB) hi = mid; else lo = mid + 1u;
    }
    const float v = __uint_as_float(hi);

    // val = sum_{d2<v} w*d2 + v*(wB - sum_{d2<v} w)   (identical to sorted form)
    float S = 0.f, Wlt = 0.f;
#pragma unroll
    for (int i = 0; i < CHUNKS; ++i) {
      const float4 d = rowD4[lane + 32 * i];
      if (d.x < v) { S = fmaf(wr[i].x, d.x, S); Wlt += wr[i].x; }
      if (d.y < v) { S = fmaf(wr[i].y, d.y, S); Wlt += wr[i].y; }
      if (d.z < v) { S = fmaf(wr[i].z, d.z, S); Wlt += wr[i].z; }
      if (d.w < v) { S = fmaf(wr[i].w, d.w, S); Wlt += wr[i].w; }
    }
    S = wave_sum(S); Wlt = wave_sum(Wlt);
    if (lane == 0)
      out[b * N_GRID + grow2] = sqrtf((S + v * (wB - Wlt)) / wB);
  }
}

extern "C" void kernel_launch(void* const* d_in, const int* in_sizes, int n_in,
                              void* d_out, int out_size, void* d_ws, size_t ws_size,
                              hipStream_t stream) {
  (void)n_in; (void)out_size; (void)d_ws; (void)ws_size;
  const float* pts  = (const float*)d_in[0];   // [B, M, 2] f32
  const float* wgt  = (const float*)d_in[1];   // [B, M]    f32
  const float* grid = (const float*)d_in[2];   // [N, 2]    f32
  float* out = (float*)d_out;                  // [B, N]    f32

  const int B = in_sizes[1] / M_PTS;           // = 2
  const int nGridBlocks = (N_GRID + 15) / 16;  // 411

  const size_t shmem = (size_t)(16 * ROWSTRIDE + M_PTS) * sizeof(float); // 140288 B
  (void)hipFuncSetAttribute((const void*)dtm_kernel,
                            hipFuncAttributeMaxDynamicSharedMemorySize,
                            (int)shmem);

  dim3 g(nGridBlocks, B);
  dtm_kernel<<<g, BLOCK, shmem, stream>>>(pts, wgt, grid, out);
}